// rollout_GNN_GRU_16707422781833
// MI455X (gfx1250) — compile-verified
//
#include <hip/hip_runtime.h>
#include <hip/hip_bf16.h>

typedef __attribute__((ext_vector_type(2))) float v2f;
typedef __attribute__((ext_vector_type(8))) float v8f;

#define N_NODES 10000
#define HDIM 128
#define DDIM 16
#define GDIM 128
#define G3 384  // 3*HDIM

__device__ __forceinline__ v2f ld2(const float* p) { return *(const v2f*)p; }

__device__ __forceinline__ v8f wmma4(v2f a, v2f b, v8f c) {
    return __builtin_amdgcn_wmma_f32_16x16x4_f32(false, a, false, b,
                                                 (short)0, c, false, false);
}

// Multi-tile streaming WMMA pass: one A panel feeds NT independent 16x16
// accumulators (NT column tiles).  Fragments for step k+1 are loaded before
// the WMMAs of step k issue (software double-buffering).
template <int K, int NT>
__device__ __forceinline__ void wmma_pass(const float* __restrict__ a_base,
                                          const float* const (&b_base)[NT],
                                          v8f (&acc)[NT])
{
    v2f a = ld2(a_base);
    v2f b[NT];
#pragma unroll
    for (int j = 0; j < NT; ++j) b[j] = ld2(b_base[j]);

#pragma unroll
    for (int k0 = 0; k0 < K; k0 += 4) {
        v2f an = a;
        v2f bn[NT];
#pragma unroll
        for (int j = 0; j < NT; ++j) bn[j] = b[j];
        if (k0 + 4 < K) {  // compile-time after full unroll
            an = ld2(a_base + k0 + 4);
#pragma unroll
            for (int j = 0; j < NT; ++j) bn[j] = ld2(b_base[j] + k0 + 4);
        }
#pragma unroll
        for (int j = 0; j < NT; ++j) acc[j] = wmma4(a, b[j], acc[j]);
        a = an;
#pragma unroll
        for (int j = 0; j < NT; ++j) b[j] = bn[j];
    }
}

// ---------------------------------------------------------------------------
// GRU gate pre-activation GEMM:
//   S  [N,384] = temp@W_ih^T + b_ih + hidden@W_hh^T + b_hh
//   Hn [N,128] = (hidden@W_hh^T + b_hh) for n-gate columns (256..383)
// One wave computes a 16x64 strip (4 tiles, 8 accumulators incl. gi/gh split).
// ---------------------------------------------------------------------------
__global__ __launch_bounds__(32) void k_gru_gemm(
    const float* __restrict__ temp,    // [N,16]
    const float* __restrict__ hidden,  // [N,128]
    const float* __restrict__ W_ih,    // [384,16]
    const float* __restrict__ W_hh,    // [384,128]
    const float* __restrict__ b_ih,    // [384]
    const float* __restrict__ b_hh,    // [384]
    float* __restrict__ S,             // [N,384]
    float* __restrict__ Hn)            // [N,128]
{
    const int lane = threadIdx.x;
    const int r0   = blockIdx.x << 4;
    const int c0   = blockIdx.y << 6;        // 64-col strip base
    const int col  = lane & 15;
    const int koff = (lane >> 4) << 1;       // 0 or 2
    const int rsel = (lane >> 4) << 3;       // 0 or 8

    int wcol[4];
    v8f acc_i[4], acc_h[4];
#pragma unroll
    for (int j = 0; j < 4; ++j) {
        wcol[j] = c0 + (j << 4) + col;
        const float bi = b_ih[wcol[j]];
        const float bh = b_hh[wcol[j]];
        acc_i[j] = (v8f){bi, bi, bi, bi, bi, bi, bi, bi};
        acc_h[j] = (v8f){bh, bh, bh, bh, bh, bh, bh, bh};
    }

    // gi: K=16 over temp
    {
        const float* a_base = temp + (r0 + col) * DDIM + koff;
        const float* b_base[4];
#pragma unroll
        for (int j = 0; j < 4; ++j) b_base[j] = W_ih + wcol[j] * DDIM + koff;
        wmma_pass<DDIM, 4>(a_base, b_base, acc_i);
    }
    // gh: K=128 over hidden
    {
        const float* a_base = hidden + (r0 + col) * HDIM + koff;
        const float* b_base[4];
#pragma unroll
        for (int j = 0; j < 4; ++j) b_base[j] = W_hh + wcol[j] * HDIM + koff;
        wmma_pass<HDIM, 4>(a_base, b_base, acc_h);
    }

    const bool is_n = (c0 >= 2 * HDIM);  // whole strip is in n-gate range or not
#pragma unroll
    for (int j = 0; j < 4; ++j) {
#pragma unroll
        for (int v = 0; v < 8; ++v) {
            const int row = r0 + rsel + v;
            const float gh = acc_h[j][v];
            S[row * G3 + wcol[j]] = acc_i[j][v] + gh;
            if (is_n) Hn[row * HDIM + (wcol[j] - 2 * HDIM)] = gh;
        }
    }
}

// ---------------------------------------------------------------------------
// GRU gate nonlinearity: hidden = (1-z)*n + z*h
// ---------------------------------------------------------------------------
__global__ void k_gates(const float* __restrict__ S, const float* __restrict__ Hn,
                        float* __restrict__ hidden)
{
    const int i = blockIdx.x * blockDim.x + threadIdx.x;
    if (i >= N_NODES * HDIM) return;
    const int row = i >> 7;
    const int j   = i & (HDIM - 1);
    const float* s = S + row * G3;
    const float sr = s[j];
    const float sz = s[HDIM + j];
    const float sn = s[2 * HDIM + j];
    const float hn = Hn[i];
    const float r = 1.0f / (1.0f + __expf(-sr));
    const float z = 1.0f / (1.0f + __expf(-sz));
    const float n = tanhf((sn - hn) + r * hn);   // i_n + r*h_n
    const float h = hidden[i];
    hidden[i] = (1.0f - z) * n + z * h;
}

__global__ void k_zero(float* __restrict__ p, int n)
{
    const int i = blockIdx.x * blockDim.x + threadIdx.x;
    if (i < n) p[i] = 0.0f;
}

__global__ void k_copy(const float* __restrict__ a, float* __restrict__ b, int n)
{
    const int i = blockIdx.x * blockDim.x + threadIdx.x;
    if (i < n) b[i] = a[i];
}

// ---------------------------------------------------------------------------
// Edge scatter-add: out[dst[e], :] += in[src[e], :]   (128-wide features)
// 32 threads per edge, float4 gather + 4 fp32 L2 atomics each.
// ---------------------------------------------------------------------------
__global__ void k_scatter(const float* __restrict__ in, const int* __restrict__ src,
                          const int* __restrict__ dst, float* __restrict__ out, int E)
{
    const int tid = blockIdx.x * blockDim.x + threadIdx.x;
    const int e = tid >> 5;
    if (e >= E) return;
    const int q = (tid & 31) << 2;
    const int s = src[e];
    const int d = dst[e];
    const float4 v = *(const float4*)(in + (size_t)s * HDIM + q);
    float* o = out + (size_t)d * HDIM + q;
    atomicAdd(o + 0, v.x);
    atomicAdd(o + 1, v.y);
    atomicAdd(o + 2, v.z);
    atomicAdd(o + 3, v.w);
}

// ---------------------------------------------------------------------------
// gnnh = tanh(agg @ W1^T + b1)   [N,128]x[128,128]; 16x64 strip per wave
// ---------------------------------------------------------------------------
__global__ __launch_bounds__(32) void k_gemm_tanh(
    const float* __restrict__ agg,  // [N,128]
    const float* __restrict__ W1,   // [128,128]
    const float* __restrict__ b1,   // [128]
    float* __restrict__ outh)       // [N,128]
{
    const int lane = threadIdx.x;
    const int r0   = blockIdx.x << 4;
    const int c0   = blockIdx.y << 6;
    const int col  = lane & 15;
    const int koff = (lane >> 4) << 1;
    const int rsel = (lane >> 4) << 3;

    int wcol[4];
    v8f acc[4];
#pragma unroll
    for (int j = 0; j < 4; ++j) {
        wcol[j] = c0 + (j << 4) + col;
        const float bb = b1[wcol[j]];
        acc[j] = (v8f){bb, bb, bb, bb, bb, bb, bb, bb};
    }

    const float* a_base = agg + (r0 + col) * HDIM + koff;
    const float* b_base[4];
#pragma unroll
    for (int j = 0; j < 4; ++j) b_base[j] = W1 + wcol[j] * HDIM + koff;
    wmma_pass<HDIM, 4>(a_base, b_base, acc);

#pragma unroll
    for (int j = 0; j < 4; ++j) {
#pragma unroll
        for (int v = 0; v < 8; ++v) {
            const int row = r0 + rsel + v;
            outh[row * GDIM + wcol[j]] = tanhf(acc[j][v]);
        }
    }
}

// ---------------------------------------------------------------------------
// xii = agg2 @ W2^T + b2 ; temp += dt*xii ; out[step] = temp
// Single 16-col tile (D=16); split-K over 4 accumulators for WMMA ILP.
// ---------------------------------------------------------------------------
__global__ __launch_bounds__(32) void k_gemm_out(
    const float* __restrict__ agg2, // [N,128]
    const float* __restrict__ W2,   // [16,128]
    const float* __restrict__ b2,   // [16]
    const float* __restrict__ t,    // [T]
    int step,
    float* __restrict__ temp,       // [N,16] read+write
    float* __restrict__ outslice)   // [N,16]
{
    const int lane = threadIdx.x;
    const int r0   = blockIdx.x << 4;
    const int col  = lane & 15;
    const int koff = (lane >> 4) << 1;
    const int rsel = (lane >> 4) << 3;

    v8f acc4[4];
#pragma unroll
    for (int j = 0; j < 4; ++j) acc4[j] = (v8f){0, 0, 0, 0, 0, 0, 0, 0};

    const float* a_base = agg2 + (r0 + col) * GDIM + koff;
    const float* b_base = W2 + col * GDIM + koff;
#pragma unroll
    for (int k0 = 0; k0 < GDIM; k0 += 16) {
#pragma unroll
        for (int j = 0; j < 4; ++j) {
            v2f a = ld2(a_base + k0 + 4 * j);
            v2f b = ld2(b_base + k0 + 4 * j);
            acc4[j] = wmma4(a, b, acc4[j]);
        }
    }
    const v8f acc = (acc4[0] + acc4[1]) + (acc4[2] + acc4[3]);

    const float bb = b2[col];
    const float dt = t[step] - t[step - 1];
#pragma unroll
    for (int v = 0; v < 8; ++v) {
        const int row = r0 + rsel + v;
        const float nt = temp[row * DDIM + col] + dt * (acc[v] + bb);
        temp[row * DDIM + col] = nt;
        outslice[row * DDIM + col] = nt;
    }
}

// ---------------------------------------------------------------------------
extern "C" void kernel_launch(void* const* d_in, const int* in_sizes, int n_in,
                              void* d_out, int out_size, void* d_ws, size_t ws_size,
                              hipStream_t stream)
{
    (void)n_in; (void)out_size; (void)ws_size;

    const float* t    = (const float*)d_in[0];
    const float* x0   = (const float*)d_in[1];
    const int*   src  = (const int*)d_in[2];
    const int*   dst  = (const int*)d_in[3];
    const float* W_ih = (const float*)d_in[4];
    const float* W_hh = (const float*)d_in[5];
    const float* b_ih = (const float*)d_in[6];
    const float* b_hh = (const float*)d_in[7];
    const float* W1   = (const float*)d_in[8];
    const float* b1   = (const float*)d_in[9];
    const float* W2   = (const float*)d_in[10];
    const float* b2   = (const float*)d_in[11];
    float* out = (float*)d_out;

    const int N = N_NODES;
    const int E = in_sizes[2];
    const int T = in_sizes[0];

    float* ws     = (float*)d_ws;
    float* temp   = ws;  ws += N * DDIM;
    float* hidden = ws;  ws += N * HDIM;
    float* S      = ws;  ws += N * G3;
    float* Hn     = ws;  ws += N * HDIM;
    float* agg    = ws;  ws += N * HDIM;
    float* gnnh   = ws;  ws += N * GDIM;
    float* agg2   = ws;  ws += N * GDIM;

    const int nh = N * HDIM;
    const int nd = N * DDIM;
    const dim3 b32(32);
    const dim3 b256(256);
    const int gb_h  = (nh + 255) / 256;
    const int gb_d  = (nd + 255) / 256;
    const int gb_sc = (int)(((size_t)E * 32 + 255) / 256);
    const int rblk  = N / 16;  // 625

    // init: temp = x0, out[0] = x0, hidden = 0
    k_copy<<<gb_d, b256, 0, stream>>>(x0, temp, nd);
    k_copy<<<gb_d, b256, 0, stream>>>(x0, out, nd);
    k_zero<<<gb_h, b256, 0, stream>>>(hidden, nh);

    for (int step = 1; step < T; ++step) {
        // hidden = GRU(temp, hidden)
        k_gru_gemm<<<dim3(rblk, G3 / 64), b32, 0, stream>>>(
            temp, hidden, W_ih, W_hh, b_ih, b_hh, S, Hn);
        k_gates<<<gb_h, b256, 0, stream>>>(S, Hn, hidden);

        // agg = segment_sum(hidden[src], dst)
        k_zero<<<gb_h, b256, 0, stream>>>(agg, nh);
        k_scatter<<<gb_sc, b256, 0, stream>>>(hidden, src, dst, agg, E);

        // gnnh = tanh(agg @ W1^T + b1)
        k_gemm_tanh<<<dim3(rblk, GDIM / 64), b32, 0, stream>>>(agg, W1, b1, gnnh);

        // agg2 = segment_sum(gnnh[src], dst)
        k_zero<<<gb_h, b256, 0, stream>>>(agg2, nh);
        k_scatter<<<gb_sc, b256, 0, stream>>>(gnnh, src, dst, agg2, E);

        // temp += dt * (agg2 @ W2^T + b2); out[step] = temp
        k_gemm_out<<<dim3(rblk, 1), b32, 0, stream>>>(
            agg2, W2, b2, t, step, temp, out + (size_t)step * N * DDIM);
    }
}